// Splatting_75986561400905
// MI455X (gfx1250) — compile-verified
//
#include <hip/hip_runtime.h>
#include <hip/hip_bf16.h>
#include <stdint.h>

#ifndef __has_builtin
#define __has_builtin(x) 0
#endif

#define B_  16
#define C_  3
#define H_  768
#define W_  768
#define HW_ (H_ * W_)          // 589824
#define EPSV 1e-7f

typedef uint32_t u32x4 __attribute__((ext_vector_type(4)));
typedef int      i32x8 __attribute__((ext_vector_type(8)));
typedef int      i32x4 __attribute__((ext_vector_type(4)));

#if defined(__HIP_DEVICE_COMPILE__) && __has_builtin(__builtin_amdgcn_tensor_load_to_lds)
#define HAVE_TDM 1
#else
#define HAVE_TDM 0
#endif

#if HAVE_TDM
// Issue one 2-D TDM tile load: nrows rows of 256 fp32 elements, row stride = HW_
// elements, from global `gaddr` into LDS byte offset `lds_byte_off`.
// Descriptor packing per CDNA5 ISA D# group0/group1 (CDNA5 ISA).
__device__ __forceinline__ void tdm_load_rows(uint32_t lds_byte_off,
                                              const void* gaddr,
                                              uint32_t nrows) {
  uint64_t ga = (uint64_t)gaddr;
  u32x4 g0;
  g0[0] = 1u;                                             // count=1, user mode
  g0[1] = lds_byte_off;                                   // lds_addr (bytes)
  g0[2] = (uint32_t)(ga & 0xFFFFFFFFu);                   // global_addr[31:0]
  g0[3] = (uint32_t)((ga >> 32) & 0x01FFFFFFu) | (2u << 30); // addr[56:32] | type=2

  i32x8 g1;
  g1[0] = 0x00020000;            // workgroup_mask=0, data_size=2 (4B), no pad/iter
  g1[1] = (int)(256u << 16);     // tensor_dim0[15:0]=256 at bits[63:48]
  g1[2] = (int)(nrows << 16);    // tensor_dim0[31:16]=0 | tensor_dim1[15:0]=nrows
  g1[3] = (int)(256u << 16);     // tensor_dim1[31:16]=0 | tile_dim0=256
  g1[4] = (int)nrows;            // tile_dim1=nrows, tile_dim2=0
  g1[5] = (int)HW_;              // tensor_dim0_stride[31:0] = H*W elements
  g1[6] = 0;                     // stride[47:32]=0 | dim1_stride[15:0]=0
  g1[7] = 0;                     // dim1_stride[47:16]=0

  i32x4 z4 = {0, 0, 0, 0};
#if __clang_major__ >= 23
  i32x8 z8 = {0, 0, 0, 0, 0, 0, 0, 0};
  __builtin_amdgcn_tensor_load_to_lds(g0, g1, z4, z4, z8, 0);
#else
  __builtin_amdgcn_tensor_load_to_lds(g0, g1, z4, z4, 0);
#endif
}
#endif

// ---------------------------------------------------------------------------
// Kernel 1: vectorized zero fill (accumulators must start at 0 every call)
// ---------------------------------------------------------------------------
__global__ void __launch_bounds__(256) zero4_kernel(float4* p, unsigned n4) {
  unsigned i = blockIdx.x * 256u + threadIdx.x;
  if (i < n4) p[i] = make_float4(0.f, 0.f, 0.f, 0.f);
}

// ---------------------------------------------------------------------------
// Kernel 2: softmax-splat scatter. One block = 256 row-contiguous source
// pixels of one batch image. Inputs staged into LDS via TDM (async tensor
// DMA), accumulation via native no-return fp32 global atomics (L2-resident:
// 151 MB accumulator < 192 MB L2).
// ---------------------------------------------------------------------------
__global__ void __launch_bounds__(256) splat_kernel(const float* __restrict__ frame,
                                                    const float* __restrict__ flow,
                                                    const float* __restrict__ imp,
                                                    float* __restrict__ out,
                                                    float* __restrict__ wsum) {
  __shared__ float smem[1536];  // [0,768): frame c0..c2 | [768,1280): dx,dy | [1280,1536): imp

  const int p0  = blockIdx.x * 256;     // flat pixel chunk start in [0, B*H*W)
  const int b   = p0 / HW_;
  const int hw0 = p0 - b * HW_;

#if HAVE_TDM
  if (threadIdx.x == 0) {
    const uint32_t lds0 = (uint32_t)(uintptr_t)(&smem[0]);  // LDS byte address (low 32b of flat)
    tdm_load_rows(lds0,            frame + (size_t)b * 3 * HW_ + hw0, 3);
    tdm_load_rows(lds0 + 768 * 4,  flow  + (size_t)b * 2 * HW_ + hw0, 2);
    tdm_load_rows(lds0 + 1280 * 4, imp   + (size_t)b * HW_     + hw0, 1);
    __builtin_amdgcn_s_wait_tensorcnt(0);
  }
  __syncthreads();
#else
  {
    const int t = threadIdx.x;
    smem[t]        = frame[(size_t)b * 3 * HW_ + hw0 + t];
    smem[256 + t]  = frame[(size_t)b * 3 * HW_ + HW_ + hw0 + t];
    smem[512 + t]  = frame[(size_t)b * 3 * HW_ + 2 * HW_ + hw0 + t];
    smem[768 + t]  = flow[(size_t)b * 2 * HW_ + hw0 + t];
    smem[1024 + t] = flow[(size_t)b * 2 * HW_ + HW_ + hw0 + t];
    smem[1280 + t] = imp[(size_t)b * HW_ + hw0 + t];
    __syncthreads();
  }
#endif

  const int t  = threadIdx.x;
  const int hw = hw0 + t;
  const int x  = hw % W_;
  const int y  = hw / W_;

  const float dx = smem[768 + t];
  const float dy = smem[1024 + t];
  const float e  = __expf(smem[1280 + t]);
  const float f0 = smem[t] * e;
  const float f1 = smem[256 + t] * e;
  const float f2 = smem[512 + t] * e;

  const float fx  = (float)x + dx;
  const float fy  = (float)y + dy;
  const float x0f = floorf(fx);
  const float y0f = floorf(fy);
  const int   x0  = (int)x0f;
  const int   y0  = (int)y0f;
  const float wx1 = fx - x0f, wx0 = 1.f - wx1;
  const float wy1 = fy - y0f, wy0 = 1.f - wy1;

  float* outb = out  + (size_t)b * 3 * HW_;
  float* wb   = wsum + (size_t)b * HW_;

#pragma unroll
  for (int cy = 0; cy < 2; ++cy) {
    const int yi = y0 + cy;
    if (yi < 0 || yi >= H_) continue;
    const float wy = cy ? wy1 : wy0;
#pragma unroll
    for (int cx = 0; cx < 2; ++cx) {
      const int xi = x0 + cx;
      if (xi < 0 || xi >= W_) continue;
      const float w  = wy * (cx ? wx1 : wx0);
      const int   di = yi * W_ + xi;
      unsafeAtomicAdd(outb + di,            f0 * w);
      unsafeAtomicAdd(outb + HW_ + di,      f1 * w);
      unsafeAtomicAdd(outb + 2 * HW_ + di,  f2 * w);
      unsafeAtomicAdd(wb + di,              e * w);
    }
  }
}

// ---------------------------------------------------------------------------
// Kernel 3: in-place normalize out /= (w + eps), float4 vectorized.
// ---------------------------------------------------------------------------
__global__ void __launch_bounds__(256) norm_kernel(float* __restrict__ out,
                                                   const float* __restrict__ wsum) {
  const size_t i = ((size_t)blockIdx.x * 256 + threadIdx.x) * 4;  // flat float idx
  const size_t chw = (size_t)3 * HW_;
  const int    b   = (int)(i / chw);
  const size_t pix = (i - (size_t)b * chw) % HW_;

  const float4 w4 = *(const float4*)(wsum + (size_t)b * HW_ + pix);
  float4 o = *(float4*)(out + i);
  o.x = o.x / (w4.x + EPSV);
  o.y = o.y / (w4.y + EPSV);
  o.z = o.z / (w4.z + EPSV);
  o.w = o.w / (w4.w + EPSV);
  *(float4*)(out + i) = o;
}

// ---------------------------------------------------------------------------
extern "C" void kernel_launch(void* const* d_in, const int* in_sizes, int n_in,
                              void* d_out, int out_size, void* d_ws, size_t ws_size,
                              hipStream_t stream) {
  const float* frame = (const float*)d_in[0];  // [16,3,768,768]
  const float* flow  = (const float*)d_in[1];  // [16,2,768,768]
  const float* imp   = (const float*)d_in[2];  // [16,1,768,768]
  float* out  = (float*)d_out;                 // [16,3,768,768]
  float* wsum = (float*)d_ws;                  // [16,768,768] weight accumulator

  const unsigned out_f4 = (B_ * C_ * HW_) / 4;  // 7,077,888
  const unsigned ws_f4  = (B_ * HW_) / 4;       // 2,359,296

  zero4_kernel<<<out_f4 / 256, 256, 0, stream>>>((float4*)out, out_f4);
  zero4_kernel<<<ws_f4 / 256, 256, 0, stream>>>((float4*)wsum, ws_f4);

  splat_kernel<<<(B_ * HW_) / 256, 256, 0, stream>>>(frame, flow, imp, out, wsum);

  norm_kernel<<<out_f4 / 256, 256, 0, stream>>>(out, wsum);
}